// SelfAttention_4148938408529
// MI455X (gfx1250) — compile-verified
//
#include <hip/hip_runtime.h>
#include <math.h>

// Problem sizes (compile-time constants from the reference)
#define BB   64
#define TT   2048
#define HH   512
#define UU   128

typedef __attribute__((ext_vector_type(2))) float v2f;
typedef __attribute__((ext_vector_type(8))) float v8f;

// ---------------------------------------------------------------------------
// Kernel 1: v[b,h] = sum_k W1[h,k] * hs[b, T-1, k]     (M=512, N=64, K=512)
// One wave per 16x16 output tile, V_WMMA_F32_16X16X4_F32, K-loop step 4.
// A (16x4 f32): lane<16 -> row M=lane, VGPR0=K0, VGPR1=K0+1; lane>=16 -> K0+2,K0+3
// B (4x16 f32): VGPR0 = row K0 (lanes 0-15) / K0+2 (lanes 16-31); VGPR1 = K0+1 / K0+3
// C/D (16x16): lane%16 = N col, VGPR r = row m0 + r + 8*(lane>=16)
// ---------------------------------------------------------------------------
__global__ __launch_bounds__(256) void v_gemm_kernel(const float* __restrict__ hs,
                                                     const float* __restrict__ W1,
                                                     float* __restrict__ v_out) {
  const int gw   = blockIdx.x * 8 + (threadIdx.x >> 5);   // 0..127 global wave
  const int lane = threadIdx.x & 31;
  const int l15  = lane & 15;
  const int hi   = lane >> 4;                             // selects K pair 0 / +2
  const int mt   = gw >> 2;                               // 32 M tiles (h)
  const int nt   = gw & 3;                                // 4 N tiles (b)
  const int m0   = mt * 16, n0 = nt * 16;

  const float* arow = W1 + (size_t)(m0 + l15) * HH + 2 * hi;                  // A[m, k..k+1]
  const float* brow = hs + ((size_t)(n0 + l15) * TT + (TT - 1)) * HH + 2 * hi; // B[k..k+1, n] = h_t

  v8f acc = {};
  for (int k0 = 0; k0 < HH; k0 += 4) {
    v2f a  = *(const v2f*)(arow + k0);
    v2f bb = *(const v2f*)(brow + k0);
    acc = __builtin_amdgcn_wmma_f32_16x16x4_f32(false, a, false, bb, (short)0, acc, false, false);
  }
  const int bcol  = n0 + l15;
  const int hbase = m0 + hi * 8;
#pragma unroll
  for (int r = 0; r < 8; ++r)
    v_out[(size_t)bcol * HH + hbase + r] = acc[r];
}

// ---------------------------------------------------------------------------
// Kernel 2: fused chunked score + local softmax + weighted-context partials.
// Block = 256 threads (8 waves), one (b, 256-row chunk) per block.
// Phase 1 streams the chunk once from HBM (coalesced float4, wave reduction),
// Phase 3 re-reads it (L2-hot, 512 KB/chunk vs 192 MB L2).
// ---------------------------------------------------------------------------
__global__ __launch_bounds__(256) void fused_score_ctx_kernel(
    const float* __restrict__ hs, const float* __restrict__ v,
    float* __restrict__ score_out, float* __restrict__ cstats,
    float* __restrict__ ctxp) {
  const int b     = blockIdx.x >> 3;
  const int chunk = blockIdx.x & 7;
  const int t0    = chunk * 256;
  const int tid   = threadIdx.x;
  const int wave  = tid >> 5, lane = tid & 31;

  __shared__ float sh_e[256];
  __shared__ float sh_red[8];

  const float* base = hs + ((size_t)b * TT + t0) * HH;
  const float* vb   = v + (size_t)b * HH;

  // -------- phase 1: scores for 256 rows (one row per wave per step) --------
  for (int r = 0; r < 32; ++r) {
    const int tt = wave + r * 8;
    const float* row = base + (size_t)tt * HH;
    float s = 0.f;
#pragma unroll
    for (int i = 0; i < 4; ++i) {
      const float4 hv = *(const float4*)(row + i * 128 + lane * 4);
      const float4 vv = *(const float4*)(vb  + i * 128 + lane * 4);
      s = fmaf(hv.x, vv.x, s); s = fmaf(hv.y, vv.y, s);
      s = fmaf(hv.z, vv.z, s); s = fmaf(hv.w, vv.w, s);
    }
#pragma unroll
    for (int off = 16; off; off >>= 1) s += __shfl_xor(s, off, 32);
    if (lane == 0) {
      sh_e[tt] = s;
      score_out[(size_t)b * TT + t0 + tt] = s;
    }
  }
  __syncthreads();

  // -------- phase 2: chunk max + exp-sum --------
  const float sval = sh_e[tid];
  float m = sval;
#pragma unroll
  for (int off = 16; off; off >>= 1) m = fmaxf(m, __shfl_xor(m, off, 32));
  if (lane == 0) sh_red[wave] = m;
  __syncthreads();
  if (tid == 0) {
    float mm = sh_red[0];
    for (int i = 1; i < 8; ++i) mm = fmaxf(mm, sh_red[i]);
    sh_red[0] = mm;
  }
  __syncthreads();
  const float cmax = sh_red[0];
  const float e = expf(sval - cmax);
  __syncthreads();                 // everyone has read sh_red[0] before reuse
  sh_e[tid] = e;
  float ssum = e;
#pragma unroll
  for (int off = 16; off; off >>= 1) ssum += __shfl_xor(ssum, off, 32);
  if (lane == 0) sh_red[wave] = ssum;
  __syncthreads();
  if (tid == 0) {
    float S = 0.f;
    for (int i = 0; i < 8; ++i) S += sh_red[i];
    cstats[((size_t)b * 8 + chunk) * 2 + 0] = cmax;
    cstats[((size_t)b * 8 + chunk) * 2 + 1] = S;
  }
  __syncthreads();

  // -------- phase 3: chunk-partial context (unnormalized, biased by cmax) ---
  float a0 = 0.f, a1 = 0.f;
#pragma unroll 4
  for (int ttc = 0; ttc < 256; ++ttc) {
    const float ew = sh_e[ttc];
    const float* row = base + (size_t)ttc * HH;
    a0 = fmaf(row[tid], ew, a0);
    a1 = fmaf(row[tid + 256], ew, a1);
  }
  float* cp = ctxp + ((size_t)chunk * BB + b) * HH;
  cp[tid] = a0;
  cp[tid + 256] = a1;
}

// ---------------------------------------------------------------------------
// Kernel 3: per-batch global max / denominator from chunk stats (tiny).
// ---------------------------------------------------------------------------
__global__ void stats_combine_kernel(const float* __restrict__ cstats,
                                     float* __restrict__ stats2) {
  const int b = threadIdx.x;   // 64 threads
  float m = cstats[(size_t)b * 16 + 0];
  for (int c = 1; c < 8; ++c) m = fmaxf(m, cstats[((size_t)b * 8 + c) * 2]);
  float S = 0.f;
  for (int c = 0; c < 8; ++c)
    S += cstats[((size_t)b * 8 + c) * 2 + 1] * expf(cstats[((size_t)b * 8 + c) * 2] - m);
  stats2[b * 2 + 0] = m;
  stats2[b * 2 + 1] = S;
}

// ---------------------------------------------------------------------------
// Kernel 4: attention_weights[b,t] = exp(score - m*) / S    -> d_out slot
// ---------------------------------------------------------------------------
__global__ __launch_bounds__(256) void weights_kernel(const float* __restrict__ score,
                                                      const float* __restrict__ stats2,
                                                      float* __restrict__ wout) {
  const int idx = blockIdx.x * 256 + threadIdx.x;   // 131072
  const int b = idx >> 11;
  wout[idx] = expf(score[idx] - stats2[b * 2]) / stats2[b * 2 + 1];
}

// ---------------------------------------------------------------------------
// Kernel 5: combine chunk context partials with log-sum-exp rescale.
// ---------------------------------------------------------------------------
__global__ __launch_bounds__(256) void ctx_combine_kernel(const float* __restrict__ ctxp,
                                                          const float* __restrict__ cstats,
                                                          const float* __restrict__ stats2,
                                                          float* __restrict__ ctx) {
  const int idx = blockIdx.x * 256 + threadIdx.x;   // 32768
  const int b = idx >> 9;
  const int h = idx & (HH - 1);
  const float m = stats2[b * 2], S = stats2[b * 2 + 1];
  float acc = 0.f;
#pragma unroll
  for (int c = 0; c < 8; ++c) {
    const float mc = cstats[((size_t)b * 8 + c) * 2];
    acc = fmaf(ctxp[((size_t)c * BB + b) * HH + h], expf(mc - m), acc);
  }
  ctx[idx] = acc / S;
}

// ---------------------------------------------------------------------------
// Kernel 6: attn_vec = tanh([ctx | h_t] @ W2)   (M=64, N=128, K=1024), WMMA f32.
// ---------------------------------------------------------------------------
__global__ __launch_bounds__(256) void final_gemm_kernel(const float* __restrict__ hs,
                                                         const float* __restrict__ ctx,
                                                         const float* __restrict__ W2,
                                                         float* __restrict__ out) {
  const int gw   = blockIdx.x * 8 + (threadIdx.x >> 5);   // 0..31
  const int lane = threadIdx.x & 31;
  const int l15  = lane & 15;
  const int hi   = lane >> 4;
  const int mt   = gw >> 3;                               // 4 M tiles (b)
  const int nt   = gw & 7;                                // 8 N tiles (units)
  const int m0   = mt * 16, n0 = nt * 16;
  const int m    = m0 + l15;
  const int ncol = n0 + l15;

  const float* actx = ctx + (size_t)m * HH;
  const float* aht  = hs + ((size_t)m * TT + (TT - 1)) * HH;

  v8f acc = {};
  for (int k0 = 0; k0 < 2 * HH; k0 += 4) {
    const int k = k0 + 2 * hi;
    const float* asrc = (k < HH) ? (actx + k) : (aht + (k - HH));
    v2f a  = *(const v2f*)asrc;
    v2f bb = { W2[(size_t)k * UU + ncol], W2[(size_t)(k + 1) * UU + ncol] };
    acc = __builtin_amdgcn_wmma_f32_16x16x4_f32(false, a, false, bb, (short)0, acc, false, false);
  }
#pragma unroll
  for (int r = 0; r < 8; ++r)
    out[(size_t)(m0 + hi * 8 + r) * UU + ncol] = tanhf(acc[r]);
}

// ---------------------------------------------------------------------------
extern "C" void kernel_launch(void* const* d_in, const int* in_sizes, int n_in,
                              void* d_out, int out_size, void* d_ws, size_t ws_size,
                              hipStream_t stream) {
  (void)in_sizes; (void)n_in; (void)out_size; (void)ws_size;
  const float* hs = (const float*)d_in[0];   // [64, 2048, 512]
  const float* W1 = (const float*)d_in[1];   // [512, 512]
  const float* W2 = (const float*)d_in[2];   // [1024, 128]
  float* out = (float*)d_out;                // [64*128] attn_vec ++ [64*2048] weights

  // workspace layout (floats)
  float* ws     = (float*)d_ws;
  float* v      = ws;                                  // 64*512      = 32768
  float* score  = v + (size_t)BB * HH;                 // 64*2048     = 131072
  float* cstats = score + (size_t)BB * TT;             // 64*8*2      = 1024
  float* ctxp   = cstats + (size_t)BB * 8 * 2;         // 8*64*512    = 262144
  float* stats2 = ctxp + (size_t)8 * BB * HH;          // 64*2        = 128
  float* ctx    = stats2 + (size_t)BB * 2;             // 64*512      = 32768

  float* weights_out = out + (size_t)BB * UU;

  // 1) v = W1 @ h_t^T          (WMMA fp32)
  v_gemm_kernel<<<16, 256, 0, stream>>>(hs, W1, v);
  // 2) fused score + chunk softmax + chunk context (single HBM pass over hs)
  fused_score_ctx_kernel<<<BB * 8, 256, 0, stream>>>(hs, v, score, cstats, ctxp);
  // 3) merge chunk stats
  stats_combine_kernel<<<1, 64, 0, stream>>>(cstats, stats2);
  // 4) emit attention_weights
  weights_kernel<<<(BB * TT) / 256, 256, 0, stream>>>(score, stats2, weights_out);
  // 5) merge chunk context partials
  ctx_combine_kernel<<<(BB * HH) / 256, 256, 0, stream>>>(ctxp, cstats, stats2, ctx);
  // 6) attn_vec = tanh([ctx | h_t] @ W2)   (WMMA fp32)
  final_gemm_kernel<<<4, 256, 0, stream>>>(hs, ctx, W2, out);
}